// SU3_CNF_53051436040473
// MI455X (gfx1250) — compile-verified
//
#include <hip/hip_runtime.h>

typedef float v2f __attribute__((ext_vector_type(2)));
typedef float v8f __attribute__((ext_vector_type(8)));

#define NSTEPS 3

// ---------- small complex helpers ----------
struct C2 { float x, y; };
__device__ __forceinline__ C2 mkc(float a, float b){ C2 r; r.x=a; r.y=b; return r; }
__device__ __forceinline__ C2 cadd(C2 a, C2 b){ return mkc(a.x+b.x, a.y+b.y); }
__device__ __forceinline__ C2 csub(C2 a, C2 b){ return mkc(a.x-b.x, a.y-b.y); }
__device__ __forceinline__ C2 cmul(C2 a, C2 b){ return mkc(a.x*b.x - a.y*b.y, a.x*b.y + a.y*b.x); }
__device__ __forceinline__ C2 cjg(C2 a){ return mkc(a.x, -a.y); }
__device__ __forceinline__ C2 cscl(C2 a, float s){ return mkc(a.x*s, a.y*s); }
__device__ __forceinline__ float cabs2(C2 a){ return a.x*a.x + a.y*a.y; }
__device__ __forceinline__ C2 cdivc(C2 a, C2 b){
  float d = cabs2(b) + 1e-30f;
  return mkc((a.x*b.x + a.y*b.y)/d, (a.y*b.x - a.x*b.y)/d);
}
__device__ __forceinline__ C2 csqrtc(C2 a){
  float m = sqrtf(sqrtf(cabs2(a)));
  float th = 0.5f*atan2f(a.y, a.x);
  return mkc(m*cosf(th), m*sinf(th));
}
__device__ __forceinline__ C2 ccbrtc(C2 a){
  float m = cbrtf(sqrtf(cabs2(a)));
  float th = atan2f(a.y, a.x)*(1.0f/3.0f);
  return mkc(m*cosf(th), m*sinf(th));
}
__device__ __forceinline__ v8f zero8(){ v8f r = {0.f,0.f,0.f,0.f,0.f,0.f,0.f,0.f}; return r; }

// LDS layout (floats)
#define OFF_W2B   0
#define OFF_W2T   4096
#define OFF_W1    8192
#define OFF_B1    8384
#define OFF_B2    8448
#define OFF_W3    8512
#define OFF_FE    8576
#define OFF_OUT   9344
#define OFF_WSC   11648
#define LDS_FLOATS 28032

__global__ __launch_bounds__(128)
void su3_cnf_kernel(const float* __restrict__ zre, const float* __restrict__ zim,
                    const float* __restrict__ gW1, const float* __restrict__ gb1,
                    const float* __restrict__ gW2, const float* __restrict__ gb2,
                    const float* __restrict__ gW3,
                    float* __restrict__ out, int B)
{
  extern __shared__ float smem[];
  float* sW2B = smem + OFF_W2B;   // W2 in B-layout:  [(j*4+n)*32 + lane]*2 + v
  float* sW2T = smem + OFF_W2T;   // W2^T in B-layout
  float* sW1  = smem + OFF_W1;    // [3][64]
  float* sb1  = smem + OFF_B1;
  float* sb2  = smem + OFF_B2;
  float* sw3  = smem + OFF_W3;
  float* sfe  = smem + OFF_FE;    // [384 rows][2] eigen re,im
  float* sout = smem + OFF_OUT;   // [384 rows][6] p,q,H00,H10,H01,H11

  const int tid  = threadIdx.x;
  const int wave = tid >> 5;
  const int lane = tid & 31;
  const int lo   = lane & 15;
  const int hi   = lane >> 4;

  // ---- build B-layout weight tables ----
  for (int e = tid; e < 4096; e += 128) {
    int v = e & 1, l = (e >> 1) & 31, blk = e >> 6;
    int n = blk & 3, j = blk >> 2;
    int K = 4*j + v + 2*(l >> 4);
    int N = 16*n + (l & 15);
    sW2B[e] = gW2[K*64 + N];
    sW2T[e] = gW2[N*64 + K];
  }
  for (int e = tid; e < 192; e += 128) sW1[e] = gW1[e];
  if (tid < 64) { sb1[tid] = gb1[tid]; sb2[tid] = gb2[tid]; sw3[tid] = gW3[tid]; }
  __syncthreads();

  float* h1buf = smem + OFF_WSC + wave*4096;  // [16][64]
  float* ubuf  = h1buf + 1024;                // [16][64]
  float* wbuf  = ubuf  + 1024;                // [16][64]
  float* stg   = wbuf  + 1024;                // [16][64] A-layout staging

  const int samp = blockIdx.x * 128 + tid;
  const int sidx = (samp < B) ? samp : (B - 1);

  C2 Y[9];
  #pragma unroll
  for (int i = 0; i < 9; ++i) Y[i] = mkc(zre[sidx*9 + i], zim[sidx*9 + i]);
  C2 divsum = mkc(0.f, 0.f);

  #pragma unroll 1
  for (int step = 0; step < NSTEPS; ++step) {
    const float tstep = (float)step * (1.0f/NSTEPS);

    // ================= scalar phase A: eigenvalues via Cardano =================
    C2 c = cadd(cadd(Y[0], Y[4]), Y[8]);
    C2 lam[3];
    {
      C2 cc = cmul(c, c), c3 = cmul(cc, c);
      C2 p  = csub(cjg(c), cscl(cc, 1.0f/3.0f));
      C2 q  = csub(mkc(-1.0f + cabs2(c)*(1.0f/3.0f), 0.f), cscl(c3, 2.0f/27.0f));
      C2 qh = cscl(q, 0.5f);
      C2 p3 = cscl(p, 1.0f/3.0f);
      C2 D  = cadd(cmul(qh, qh), cmul(cmul(p3, p3), p3));
      C2 u  = ccbrtc(cadd(cscl(q, -0.5f), csqrtc(D)));
      if (cabs2(u) < 1e-16f) u = mkc(1e-8f, 0.f);
      C2 vv = cscl(cdivc(p, u), -1.0f/3.0f);
      C2 w  = mkc(-0.5f, 0.86602540378f);
      C2 c3rd = cscl(c, 1.0f/3.0f);
      lam[0] = cadd(cadd(u, vv), c3rd);
      lam[1] = cadd(cadd(cmul(u, w), cmul(vv, cjg(w))), c3rd);
      lam[2] = cadd(cadd(cmul(u, cjg(w)), cmul(vv, w)), c3rd);
    }
    #pragma unroll
    for (int k = 0; k < 3; ++k) {
      int row = wave*96 + lane*3 + k;
      sfe[row*2 + 0] = lam[k].x;
      sfe[row*2 + 1] = lam[k].y;
    }
    __syncthreads();

    // ================= MLP phase: 6 tiles of 16 rows per wave =================
    #pragma unroll 1
    for (int tt = 0; tt < 6; ++tt) {
      const int rowbase = wave*96 + tt*16;
      const float fRe = sfe[(rowbase + lo)*2 + 0];
      const float fIm = sfe[(rowbase + lo)*2 + 1];

      // h1 = tanh(feats@W1 + b1), built directly in fp32 A-layout
      v2f aA[16];
      #pragma unroll
      for (int j = 0; j < 16; ++j) {
        int K0 = 4*j + 2*hi;
        float z0 = fRe*sW1[K0]   + fIm*sW1[64+K0]   + tstep*sW1[128+K0]   + sb1[K0];
        float z1 = fRe*sW1[K0+1] + fIm*sW1[64+K0+1] + tstep*sW1[128+K0+1] + sb1[K0+1];
        float h0 = tanhf(z0), h1v = tanhf(z1);
        aA[j].x = h0; aA[j].y = h1v;
        *(v2f*)&h1buf[lo*64 + K0] = aA[j];
      }

      // ---- forward GEMM: z2 = h1 @ W2 ----
      v8f acc[4];
      #pragma unroll
      for (int n = 0; n < 4; ++n) acc[n] = zero8();
      #pragma unroll
      for (int j = 0; j < 16; ++j) {
        v2f av = aA[j];
        #pragma unroll
        for (int n = 0; n < 4; ++n) {
          v2f bv = *(const v2f*)&sW2B[(((j<<2) + n)*32 + lane)*2];
          acc[n] = __builtin_amdgcn_wmma_f32_16x16x4_f32(false, av, false, bv, (short)0, acc[n], false, false);
        }
      }
      // h2 = tanh(z2 + b2); v2 = (1-h2^2)*w3 -> stage in (row,col) layout
      #pragma unroll
      for (int n = 0; n < 4; ++n) {
        int col = 16*n + lo;
        float b2c = sb2[col], w3c = sw3[col];
        #pragma unroll
        for (int r = 0; r < 8; ++r) {
          float h2 = tanhf(acc[n][r] + b2c);
          acc[n][r] = h2;                       // keep h2 live in acc
          stg[(r + 8*hi)*64 + col] = (1.f - h2*h2) * w3c;
        }
      }
      // ---- backward GEMM: u = v2 @ W2^T ----
      v8f au[4];
      #pragma unroll
      for (int n = 0; n < 4; ++n) au[n] = zero8();
      #pragma unroll
      for (int j = 0; j < 16; ++j) {
        v2f av = *(const v2f*)&stg[lo*64 + 4*j + 2*hi];
        #pragma unroll
        for (int n = 0; n < 4; ++n) {
          v2f bv = *(const v2f*)&sW2T[(((j<<2) + n)*32 + lane)*2];
          au[n] = __builtin_amdgcn_wmma_f32_16x16x4_f32(false, av, false, bv, (short)0, au[n], false, false);
        }
      }
      #pragma unroll
      for (int n = 0; n < 4; ++n)
        #pragma unroll
        for (int r = 0; r < 8; ++r)
          ubuf[(r + 8*hi)*64 + 16*n + lo] = au[n][r];

      // gradient reduction: p = W1[0,:] . v1, q = W1[1,:] . v1, v1 = (1-h1^2)*u
      if (lane < 16) {
        float p = 0.f, q = 0.f;
        for (int k = 0; k < 64; ++k) {
          float h  = h1buf[lane*64 + k];
          float v1 = (1.f - h*h) * ubuf[lane*64 + k];
          p += sW1[k]*v1; q += sW1[64+k]*v1;
        }
        sout[(rowbase + lane)*6 + 0] = p;
        sout[(rowbase + lane)*6 + 1] = q;
      }

      // ---- two Hessian tangent passes (dirs: d/dReL, d/dImL) ----
      #pragma unroll 1
      for (int e = 0; e < 2; ++e) {
        v8f at[4];
        #pragma unroll
        for (int n = 0; n < 4; ++n) at[n] = zero8();
        #pragma unroll
        for (int j = 0; j < 16; ++j) {
          int K0 = 4*j + 2*hi;
          v2f da;
          da.x = (1.f - aA[j].x*aA[j].x) * sW1[e*64 + K0];
          da.y = (1.f - aA[j].y*aA[j].y) * sW1[e*64 + K0 + 1];
          #pragma unroll
          for (int n = 0; n < 4; ++n) {
            v2f bv = *(const v2f*)&sW2B[(((j<<2) + n)*32 + lane)*2];
            at[n] = __builtin_amdgcn_wmma_f32_16x16x4_f32(false, da, false, bv, (short)0, at[n], false, false);
          }
        }
        // dv2 = -2*h2*dh2*w3, dh2 = (1-h2^2)*dz2
        #pragma unroll
        for (int n = 0; n < 4; ++n) {
          int col = 16*n + lo;
          float w3c = sw3[col];
          #pragma unroll
          for (int r = 0; r < 8; ++r) {
            float h2  = acc[n][r];
            float dh2 = (1.f - h2*h2) * at[n][r];
            stg[(r + 8*hi)*64 + col] = -2.f * h2 * dh2 * w3c;
          }
        }
        v8f aw[4];
        #pragma unroll
        for (int n = 0; n < 4; ++n) aw[n] = zero8();
        #pragma unroll
        for (int j = 0; j < 16; ++j) {
          v2f av = *(const v2f*)&stg[lo*64 + 4*j + 2*hi];
          #pragma unroll
          for (int n = 0; n < 4; ++n) {
            v2f bv = *(const v2f*)&sW2T[(((j<<2) + n)*32 + lane)*2];
            aw[n] = __builtin_amdgcn_wmma_f32_16x16x4_f32(false, av, false, bv, (short)0, aw[n], false, false);
          }
        }
        #pragma unroll
        for (int n = 0; n < 4; ++n)
          #pragma unroll
          for (int r = 0; r < 8; ++r)
            wbuf[(r + 8*hi)*64 + 16*n + lo] = aw[n][r];

        if (lane < 16) {
          float hp = 0.f, hq = 0.f;
          for (int k = 0; k < 64; ++k) {
            float h   = h1buf[lane*64 + k];
            float hs  = 1.f - h*h;
            float dh1 = hs * sW1[e*64 + k];
            float dv1 = -2.f*h*dh1*ubuf[lane*64 + k] + hs*wbuf[lane*64 + k];
            hp += sW1[k]*dv1; hq += sW1[64+k]*dv1;
          }
          sout[(rowbase + lane)*6 + 2 + 2*e] = hp;
          sout[(rowbase + lane)*6 + 3 + 2*e] = hq;
        }
      }
    }
    __syncthreads();

    // ================= scalar phase B: field, divergence, expmap =================
    float Pp[3], Qq[3], H00[3], H10[3], H01[3], H11[3];
    #pragma unroll
    for (int k = 0; k < 3; ++k) {
      int row = wave*96 + lane*3 + k;
      Pp[k]  = sout[row*6+0]; Qq[k]  = sout[row*6+1];
      H00[k] = sout[row*6+2]; H10[k] = sout[row*6+3];
      H01[k] = sout[row*6+4]; H11[k] = sout[row*6+5];
    }
    C2 Fc = mkc(0,0), dFcx = mkc(0,0), dFcy = mkc(0,0);
    #pragma unroll
    for (int k = 0; k < 3; ++k) {
      C2 l  = lam[k];
      C2 ll = cmul(l, l);
      C2 Dg = cadd(csub(cscl(ll, 3.f), cscl(cmul(c, l), 2.f)), cjg(c));
      if (cabs2(Dg) < 1e-12f) Dg = cadd(Dg, mkc(1e-6f, 0.f));
      C2 lc  = cdivc(ll, Dg);
      C2 lcb = cscl(cdivc(l, Dg), -1.f);
      C2 sc  = csub(cscl(l, 6.f), cscl(c, 2.f));
      C2 dDdc  = csub(cmul(sc, lc), cscl(l, 2.f));
      C2 dDdcb = cadd(cmul(sc, lcb), mkc(1.f, 0.f));
      C2 num   = csub(cscl(cmul(cmul(l, lc), Dg), 2.f), cmul(ll, dDdc));
      C2 lcc   = cdivc(cdivc(num, Dg), Dg);
      num      = csub(cscl(cmul(cmul(l, lcb), Dg), 2.f), cmul(ll, dDdcb));
      C2 lccb  = cdivc(cdivc(num, Dg), Dg);
      num      = cadd(cscl(cmul(lcb, Dg), -1.f), cmul(l, dDdcb));
      C2 lcbcb = cdivc(cdivc(num, Dg), Dg);

      C2 phl  = mkc(0.5f*Pp[k], -0.5f*Qq[k]);
      C2 phlb = mkc(0.5f*Pp[k],  0.5f*Qq[k]);
      C2 mu   = cjg(lcb);
      Fc = cadd(Fc, cadd(cmul(phl, lc), cmul(phlb, mu)));
      #pragma unroll
      for (int v = 0; v < 2; ++v) {
        C2 dc  = v ? mkc(0.f,1.f) : mkc(1.f,0.f);
        C2 dcb = cjg(dc);
        C2 dl  = cadd(cmul(lc, dc), cmul(lcb, dcb));
        float dp = H00[k]*dl.x + H01[k]*dl.y;
        float dq = H10[k]*dl.x + H11[k]*dl.y;
        C2 dphl  = mkc(0.5f*dp, -0.5f*dq);
        C2 dphlb = mkc(0.5f*dp,  0.5f*dq);
        C2 dlc   = cadd(cmul(lcc,  dc), cmul(lccb,  dcb));
        C2 dlcb  = cadd(cmul(lccb, dc), cmul(lcbcb, dcb));
        C2 term  = cadd(cadd(cmul(dphl, lc), cmul(phl, dlc)),
                        cadd(cmul(dphlb, mu), cmul(phlb, cjg(dlcb))));
        if (v == 0) dFcx = cadd(dFcx, term); else dFcy = cadd(dFcy, term);
      }
    }
    C2 s = mkc(2.f*Fc.x, -2.f*Fc.y);
    float Hf00 = 2.f*dFcx.x, Hf10 = -2.f*dFcx.y;
    float Hf01 = 2.f*dFcy.x, Hf11 = -2.f*dFcy.y;

    C2 yH[9], A[9];
    #pragma unroll
    for (int i = 0; i < 3; ++i)
      #pragma unroll
      for (int j = 0; j < 3; ++j) yH[i*3+j] = cjg(Y[j*3+i]);
    C2 trA = cscl(csub(cmul(s, cjg(c)), cmul(cjg(s), c)), 0.5f);
    #pragma unroll
    for (int i = 0; i < 3; ++i)
      #pragma unroll
      for (int j = 0; j < 3; ++j) {
        C2 a0 = cscl(csub(cmul(s, yH[i*3+j]), cmul(cjg(s), Y[i*3+j])), 0.5f);
        if (i == j) a0 = csub(a0, cscl(trA, 1.0f/3.0f));
        A[i*3+j] = a0;
      }

    // divergence: 18 analytic JVP directions
    C2 divc = mkc(0.f, 0.f);
    #pragma unroll 1
    for (int d = 0; d < 18; ++d) {
      int part = d & 1, klf = d >> 1;
      int kk = klf / 3, llx = klf % 3;
      C2 zeta = part ? mkc(0.f,1.f) : mkc(1.f,0.f);
      C2 Z[9];
      #pragma unroll
      for (int i = 0; i < 9; ++i) Z[i] = mkc(0.f,0.f);
      Z[kk*3 + llx] = zeta;
      C2 cp = (kk == llx) ? zeta : mkc(0.f,0.f);
      C2 sp = mkc(Hf00*cp.x + Hf01*cp.y, Hf10*cp.x + Hf11*cp.y);
      C2 trp = cscl(csub(cadd(cmul(sp, cjg(c)), cmul(s, cjg(cp))),
                         cadd(cmul(cjg(sp), c), cmul(cjg(s), cp))), 0.5f);
      C2 Ap[9];
      #pragma unroll
      for (int i = 0; i < 3; ++i)
        #pragma unroll
        for (int j = 0; j < 3; ++j) {
          C2 a0 = cadd(cmul(sp, yH[i*3+j]), cmul(s, cjg(Z[j*3+i])));
          a0 = csub(a0, cadd(cmul(cjg(sp), Y[i*3+j]), cmul(cjg(s), Z[i*3+j])));
          a0 = cscl(a0, 0.5f);
          if (i == j) a0 = csub(a0, cscl(trp, 1.0f/3.0f));
          Ap[i*3+j] = a0;
        }
      float U[9];
      #pragma unroll
      for (int i = 0; i < 3; ++i)
        #pragma unroll
        for (int j = 0; j < 3; ++j) {
          C2 a0 = mkc(0.f,0.f);
          #pragma unroll
          for (int m = 0; m < 3; ++m)
            a0 = cadd(a0, cadd(cmul(Z[i*3+m], A[m*3+j]), cmul(Y[i*3+m], Ap[m*3+j])));
          U[i*3+j] = a0.x;
        }
      C2 am[9], S[9];
      #pragma unroll
      for (int i = 0; i < 3; ++i)
        #pragma unroll
        for (int j = 0; j < 3; ++j) {
          C2 a0 = mkc(0.f,0.f);
          #pragma unroll
          for (int m = 0; m < 3; ++m) a0 = cadd(a0, cmul(yH[i*3+m], Z[m*3+j]));
          am[i*3+j] = a0;
        }
      #pragma unroll
      for (int i = 0; i < 3; ++i)
        #pragma unroll
        for (int j = 0; j < 3; ++j) S[i*3+j] = cscl(csub(am[i*3+j], cjg(am[j*3+i])), 0.5f);
      C2 trS = cadd(cadd(S[0], S[4]), S[8]);
      #pragma unroll
      for (int i = 0; i < 3; ++i) S[i*4] = csub(S[i*4], cscl(trS, 1.0f/3.0f));
      #pragma unroll
      for (int i = 0; i < 3; ++i)
        #pragma unroll
        for (int j = 0; j < 3; ++j) {
          C2 a0 = mkc(0.f,0.f);
          #pragma unroll
          for (int m = 0; m < 3; ++m) a0 = cadd(a0, cmul(Y[i*3+m], S[m*3+j]));
          divc.x += U[i*3+j]*a0.x;
          divc.y += U[i*3+j]*a0.y;
        }
    }
    divsum = cadd(divsum, cscl(divc, 1.0f/NSTEPS));

    // expmap: Y <- Y @ expm(A/NSTEPS) via closed-form 3x3 Hermitian eig of H=-i*A/NSTEPS
    C2 Hh[9];
    #pragma unroll
    for (int i = 0; i < 9; ++i) { C2 ae = cscl(A[i], 1.0f/NSTEPS); Hh[i] = mkc(ae.y, -ae.x); }
    #pragma unroll
    for (int i = 0; i < 3; ++i) {
      Hh[i*4].y = 0.f;
      #pragma unroll
      for (int j = 0; j < 3; ++j)
        if (j > i) {
          C2 hij = cscl(cadd(Hh[i*3+j], cjg(Hh[j*3+i])), 0.5f);
          Hh[i*3+j] = hij; Hh[j*3+i] = cjg(hij);
        }
    }
    float tr2 = 0.f;
    #pragma unroll
    for (int i = 0; i < 9; ++i) tr2 += cabs2(Hh[i]);
    C2 Ynew[9];
    if (tr2 < 1e-20f) {
      #pragma unroll
      for (int i = 0; i < 3; ++i)
        #pragma unroll
        for (int j = 0; j < 3; ++j) {
          C2 a0 = Y[i*3+j];
          #pragma unroll
          for (int m = 0; m < 3; ++m) a0 = cadd(a0, cscl(cmul(Y[i*3+m], A[m*3+j]), 1.0f/NSTEPS));
          Ynew[i*3+j] = a0;
        }
    } else {
      float pp = -0.5f*tr2;
      C2 t0 = csub(cmul(Hh[4],Hh[8]), cmul(Hh[5],Hh[7]));
      C2 t1 = csub(cmul(Hh[3],Hh[8]), cmul(Hh[5],Hh[6]));
      C2 t2 = csub(cmul(Hh[3],Hh[7]), cmul(Hh[4],Hh[6]));
      C2 det = cadd(csub(cmul(Hh[0],t0), cmul(Hh[1],t1)), cmul(Hh[2],t2));
      float qq = -det.x;
      float mmag = 2.f*sqrtf(tr2/6.f);
      float arg = 3.f*qq/(pp*mmag + copysignf(1e-30f, pp*mmag));
      arg = fminf(1.f, fmaxf(-1.f, arg));
      float phi = acosf(arg)*(1.0f/3.0f);
      float ev[3]; C2 eph[3]; C2 V[9];
      #pragma unroll
      for (int k = 0; k < 3; ++k) {
        ev[k] = mmag*cosf(phi - 2.0943951024f*(float)k);
        eph[k] = mkc(cosf(ev[k]), sinf(ev[k]));
        C2 M[9];
        #pragma unroll
        for (int i = 0; i < 9; ++i) M[i] = Hh[i];
        #pragma unroll
        for (int i = 0; i < 3; ++i) M[i*4] = csub(M[i*4], mkc(ev[k], 0.f));
        C2 cA[3], cB[3], cC[3];
        cA[0] = csub(cmul(M[1],M[5]), cmul(M[2],M[4]));
        cA[1] = csub(cmul(M[2],M[3]), cmul(M[0],M[5]));
        cA[2] = csub(cmul(M[0],M[4]), cmul(M[1],M[3]));
        cB[0] = csub(cmul(M[1],M[8]), cmul(M[2],M[7]));
        cB[1] = csub(cmul(M[2],M[6]), cmul(M[0],M[8]));
        cB[2] = csub(cmul(M[0],M[7]), cmul(M[1],M[6]));
        cC[0] = csub(cmul(M[4],M[8]), cmul(M[5],M[7]));
        cC[1] = csub(cmul(M[5],M[6]), cmul(M[3],M[8]));
        cC[2] = csub(cmul(M[3],M[7]), cmul(M[4],M[6]));
        float nA = cabs2(cA[0])+cabs2(cA[1])+cabs2(cA[2]);
        float nB = cabs2(cB[0])+cabs2(cB[1])+cabs2(cB[2]);
        float nC = cabs2(cC[0])+cabs2(cC[1])+cabs2(cC[2]);
        C2* best = cA; float nb = nA;
        if (nB > nb) { best = cB; nb = nB; }
        if (nC > nb) { best = cC; nb = nC; }
        float inv = rsqrtf(nb + 1e-30f);
        #pragma unroll
        for (int i = 0; i < 3; ++i) V[i*3+k] = cscl(best[i], inv);
      }
      C2 E[9];
      #pragma unroll
      for (int i = 0; i < 3; ++i)
        #pragma unroll
        for (int j = 0; j < 3; ++j) {
          C2 a0 = mkc(0.f,0.f);
          #pragma unroll
          for (int k = 0; k < 3; ++k)
            a0 = cadd(a0, cmul(eph[k], cmul(V[i*3+k], cjg(V[j*3+k]))));
          E[i*3+j] = a0;
        }
      #pragma unroll
      for (int i = 0; i < 3; ++i)
        #pragma unroll
        for (int j = 0; j < 3; ++j) {
          C2 a0 = mkc(0.f,0.f);
          #pragma unroll
          for (int m = 0; m < 3; ++m) a0 = cadd(a0, cmul(Y[i*3+m], E[m*3+j]));
          Ynew[i*3+j] = a0;
        }
    }
    #pragma unroll
    for (int i = 0; i < 9; ++i) Y[i] = Ynew[i];
    __syncthreads();
  }

  if (samp < B) {
    #pragma unroll
    for (int i = 0; i < 9; ++i) {
      out[(samp*9 + i)*2 + 0] = Y[i].x;
      out[(samp*9 + i)*2 + 1] = Y[i].y;
    }
    out[B*18 + samp] = divsum.x + divsum.y;
  }
}

extern "C" void kernel_launch(void* const* d_in, const int* in_sizes, int n_in,
                              void* d_out, int out_size, void* d_ws, size_t ws_size,
                              hipStream_t stream) {
  const float* zre = (const float*)d_in[0];
  const float* zim = (const float*)d_in[1];
  const float* W1  = (const float*)d_in[2];
  const float* b1  = (const float*)d_in[3];
  const float* W2  = (const float*)d_in[4];
  const float* b2  = (const float*)d_in[5];
  const float* W3  = (const float*)d_in[6];
  // d_in[7] = b3: constant offset, does not affect gradients/divergence/output.
  (void)d_ws; (void)ws_size; (void)n_in; (void)out_size;
  int B = in_sizes[0] / 9;
  int grid = (B + 127) / 128;
  size_t shbytes = LDS_FLOATS * sizeof(float);
  su3_cnf_kernel<<<grid, 128, shbytes, stream>>>(zre, zim, W1, b1, W2, b2, W3, (float*)d_out, B);
}